// GATLayer_23931557773782
// MI455X (gfx1250) — compile-verified
//
#include <hip/hip_runtime.h>
#include <math.h>

// ---------------------------------------------------------------------------
// GAT layer (heads=1, self-loops, segment softmax) for gfx1250 / MI455X.
//
// Pipeline (all fp32, matching the fp32 reference):
//   1) h = x @ W                      -- V_WMMA_F32_16X16X4_F32 tensor cores
//   2) a_src = h@att_src, a_dst = h@att_dst
//   3) per-edge logits + leaky-relu + segment max  (float atomic max trick)
//   4) e = exp(logit - max[dst]); denom[dst] += e  (global_atomic_add_f32)
//   5) out[dst] += (e/denom[dst]) * h[src]         (wave-per-edge scatter)
//   6) out = relu(out + bias)
//
// h (12.8 MB) and out (12.8 MB) both live comfortably inside the 192 MB L2,
// so the scattered edge-phase gathers/atomics (the real bottleneck, ~440 MB
// of L2 traffic) never touch HBM after the first pass.
// ---------------------------------------------------------------------------

typedef __attribute__((ext_vector_type(2))) float v2f;
typedef __attribute__((ext_vector_type(8))) float v8f;

#define F_IN      128
#define F_OUT     64
#define NEG_SLOPE 0.2f
#define GAT_EPS   1e-16f

// ---------------- stage 0: init seg_max=-inf, denom=0, out=0 ---------------
__global__ void k_init(float* __restrict__ seg_max, float* __restrict__ denom,
                       float* __restrict__ out, int nNodes) {
    int i = blockIdx.x * blockDim.x + threadIdx.x;
    if (i < nNodes) {
        seg_max[i] = -__builtin_inff();
        denom[i]   = 0.0f;
    }
    if (i < nNodes * F_OUT) out[i] = 0.0f;
}

// ---------------- stage 1: h = x @ W via fp32 WMMA -------------------------
// One wave32 per 16x16 output tile. VGPR layouts per CDNA5 ISA 7.12.2:
//   A (16x4 f32):  lane L holds row M=L%16; VGPR v holds K = 2*(L/16)+v
//   B (4x16 f32):  lane L holds col N=L%16; VGPR v holds K = 2*(L/16)+v
//   C/D (16x16):   lane L col N=L%16; VGPR r holds row M = 8*(L/16)+r
__global__ __launch_bounds__(32)
void k_gemm_h_wmma(const float* __restrict__ x, const float* __restrict__ W,
                   float* __restrict__ h, int nNodes) {
    const int mBase = blockIdx.x * 16;
    const int nBase = blockIdx.y * 16;
    const int lane  = threadIdx.x;      // 0..31 (wave32)
    const int mL    = lane & 15;
    const int kh    = lane >> 4;        // 0 or 1

    int rowA = mBase + mL;
    if (rowA >= nNodes) rowA = nNodes - 1;   // clamp reads; EXEC stays all-1s

    const float* xrow = x + (size_t)rowA * F_IN + 2 * kh;

    v8f c = {};
    #pragma unroll
    for (int k0 = 0; k0 < F_IN; k0 += 4) {
        v2f a, b;
        a.x = xrow[k0];
        a.y = xrow[k0 + 1];
        const int kb = k0 + 2 * kh;
        b.x = W[(size_t)kb       * F_OUT + nBase + mL];
        b.y = W[(size_t)(kb + 1) * F_OUT + nBase + mL];
        // D = A*B + C ; 8 args: (neg_a, A, neg_b, B, c_mod, C, reuse_a, reuse_b)
        c = __builtin_amdgcn_wmma_f32_16x16x4_f32(
                false, a, false, b, (short)0, c, false, false);
    }

    #pragma unroll
    for (int r = 0; r < 8; ++r) {
        const int row = mBase + kh * 8 + r;
        if (row < nNodes)
            h[(size_t)row * F_OUT + nBase + mL] = c[r];
    }
}

// ---------------- stage 2: a_src / a_dst projections -----------------------
__global__ void k_attn_proj(const float* __restrict__ h,
                            const float* __restrict__ att_src,
                            const float* __restrict__ att_dst,
                            float* __restrict__ a_src, float* __restrict__ a_dst,
                            int nNodes) {
    int n = blockIdx.x * blockDim.x + threadIdx.x;
    if (n >= nNodes) return;
    const float4* hp = (const float4*)(h + (size_t)n * F_OUT);
    const float4* as = (const float4*)att_src;
    const float4* ad = (const float4*)att_dst;
    float ss = 0.0f, sd = 0.0f;
    #pragma unroll
    for (int i = 0; i < F_OUT / 4; ++i) {
        float4 hv = hp[i], a = as[i], d = ad[i];
        ss += hv.x * a.x + hv.y * a.y + hv.z * a.z + hv.w * a.w;
        sd += hv.x * d.x + hv.y * d.y + hv.z * d.z + hv.w * d.w;
    }
    a_src[n] = ss;
    a_dst[n] = sd;
}

// float atomic max via sign-split int/uint atomics (handles negatives; init -inf)
__device__ __forceinline__ void atomicMaxF(float* addr, float val) {
    if (val >= 0.0f) atomicMax((int*)addr, __float_as_int(val));
    else             atomicMin((unsigned int*)addr, __float_as_uint(val));
}

// ---------------- stage 3: logits + segment max ----------------------------
// edge list layout: ei[0:E) = src, ei[E:2E) = dst; edges [E, E+N) are self loops
__global__ void k_edge_logits(const int* __restrict__ ei,
                              const float* __restrict__ a_src,
                              const float* __restrict__ a_dst,
                              float* __restrict__ logits,
                              float* __restrict__ seg_max, int E, int nNodes) {
    int e = blockIdx.x * blockDim.x + threadIdx.x;
    if (e >= E + nNodes) return;
    const int s = (e < E) ? ei[e]     : (e - E);
    const int d = (e < E) ? ei[E + e] : (e - E);
    float t = a_src[s] + a_dst[d];
    t = (t > 0.0f) ? t : NEG_SLOPE * t;
    logits[e] = t;
    atomicMaxF(&seg_max[d], t);
}

// ---------------- stage 4: exp + segment sum -------------------------------
__global__ void k_edge_exp(const int* __restrict__ ei,
                           const float* __restrict__ logits,
                           const float* __restrict__ seg_max,
                           float* __restrict__ expv, float* __restrict__ denom,
                           int E, int nNodes) {
    int e = blockIdx.x * blockDim.x + threadIdx.x;
    if (e >= E + nNodes) return;
    const int d = (e < E) ? ei[E + e] : (e - E);
    const float v = expf(logits[e] - seg_max[d]);
    expv[e] = v;
    atomicAdd(&denom[d], v);
}

// ---------------- stage 5: weighted feature scatter ------------------------
// One wave32 per edge; lane l handles features [2l, 2l+1] -> float2 gather of
// h[src] and two global_atomic_add_f32 into out[dst], coalesced per wave.
__global__ void k_scatter(const int* __restrict__ ei,
                          const float* __restrict__ expv,
                          const float* __restrict__ denom,
                          const float* __restrict__ h,
                          float* __restrict__ out, int E, int nNodes) {
    const long long tid = (long long)blockIdx.x * blockDim.x + threadIdx.x;
    const int e    = (int)(tid >> 5);
    const int lane = (int)(tid & 31);
    if (e >= E + nNodes) return;
    const int s = (e < E) ? ei[e]     : (e - E);
    const int d = (e < E) ? ei[E + e] : (e - E);
    const float alpha = expv[e] / (denom[d] + GAT_EPS);
    const float2 hv = *(const float2*)(h + (size_t)s * F_OUT + 2 * lane);
    float* op = out + (size_t)d * F_OUT + 2 * lane;
    atomicAdd(op,     alpha * hv.x);
    atomicAdd(op + 1, alpha * hv.y);
}

// ---------------- stage 6: bias + relu -------------------------------------
__global__ void k_bias_relu(float* __restrict__ out, const float* __restrict__ bias,
                            int nNodes) {
    int i = blockIdx.x * blockDim.x + threadIdx.x;
    if (i >= nNodes * F_OUT) return;
    const float v = out[i] + bias[i & (F_OUT - 1)];
    out[i] = (v > 0.0f) ? v : 0.0f;
}

// ---------------------------------------------------------------------------
extern "C" void kernel_launch(void* const* d_in, const int* in_sizes, int n_in,
                              void* d_out, int out_size, void* d_ws, size_t ws_size,
                              hipStream_t stream) {
    const float* x       = (const float*)d_in[0];
    const int*   ei      = (const int*)  d_in[1];   // int32 (JAX x64 off)
    const float* W       = (const float*)d_in[2];
    const float* att_src = (const float*)d_in[3];
    const float* att_dst = (const float*)d_in[4];
    const float* bias    = (const float*)d_in[5];
    float*       out     = (float*)d_out;

    const int nNodes = in_sizes[0] / F_IN;   // 50000
    const int E      = in_sizes[1] / 2;      // 800000
    const int total  = E + nNodes;           // edges incl. self loops

    // carve workspace (256B aligned slabs); ~20.5 MB total
    char* p = (char*)d_ws;
    auto carve = [&](size_t bytes) -> float* {
        float* r = (float*)p;
        p += (bytes + 255) & ~(size_t)255;
        return r;
    };
    float* h       = carve((size_t)nNodes * F_OUT * sizeof(float));
    float* a_src   = carve((size_t)nNodes * sizeof(float));
    float* a_dst   = carve((size_t)nNodes * sizeof(float));
    float* seg_max = carve((size_t)nNodes * sizeof(float));
    float* denom   = carve((size_t)nNodes * sizeof(float));
    float* logits  = carve((size_t)total * sizeof(float));
    float* expv    = carve((size_t)total * sizeof(float));

    const int B = 256;

    k_init<<<(nNodes * F_OUT + B - 1) / B, B, 0, stream>>>(seg_max, denom, out, nNodes);

    dim3 gemmGrid((nNodes + 15) / 16, F_OUT / 16);
    k_gemm_h_wmma<<<gemmGrid, 32, 0, stream>>>(x, W, h, nNodes);

    k_attn_proj<<<(nNodes + B - 1) / B, B, 0, stream>>>(h, att_src, att_dst,
                                                        a_src, a_dst, nNodes);

    k_edge_logits<<<(total + B - 1) / B, B, 0, stream>>>(ei, a_src, a_dst,
                                                         logits, seg_max, E, nNodes);

    k_edge_exp<<<(total + B - 1) / B, B, 0, stream>>>(ei, logits, seg_max,
                                                      expv, denom, E, nNodes);

    const long long scatterThreads = (long long)total * 32;
    k_scatter<<<(unsigned)((scatterThreads + B - 1) / B), B, 0, stream>>>(
        ei, expv, denom, h, out, E, nNodes);

    k_bias_relu<<<(nNodes * F_OUT + B - 1) / B, B, 0, stream>>>(out, bias, nNodes);
}